// QuantumCircuit_15496242004747
// MI455X (gfx1250) — compile-verified
//
#include <hip/hip_runtime.h>

typedef __attribute__((ext_vector_type(2))) float v2f;
typedef __attribute__((ext_vector_type(8))) float v8f;

#define BQ 8192

// ---------------- tiny 2x2 / 4x4 element helpers ----------------
__device__ __forceinline__ float ry_e(float ct, float st, int r, int c) {
    // RY(theta) = [[cos, -sin],[sin, cos]] of theta/2
    return (r == 0) ? ((c == 0) ? ct : -st) : ((c == 0) ? st : ct);
}
__device__ __forceinline__ float I_e(int r, int c) { return (r == c) ? 1.f : 0.f; }
__device__ __forceinline__ float X_e(int r, int c) { return (r != c) ? 1.f : 0.f; }
__device__ __forceinline__ float CNOT_e(int r, int c) {
    int p = (r < 2) ? r : (5 - r); // 2->3, 3->2
    return (c == p) ? 1.f : 0.f;
}
__device__ __forceinline__ int swap12(int x) {
    // exchange bit2 and bit1
    return (x & 0x9) | ((x & 2) << 1) | ((x & 4) >> 1);
}

// ---------------- kernel 1: build the 16x16 unitary U ----------------
__global__ void build_U(const float* __restrict__ t1, const float* __restrict__ t2,
                        const float* __restrict__ t3, const float* __restrict__ t4,
                        const float* __restrict__ t5, const float* __restrict__ t6,
                        float* __restrict__ U) {
    __shared__ float A[16][16], Bm[16][16], Cm[16][16];
    const int t = threadIdx.x;
    const int r = t >> 4, c = t & 15;

    const float c1 = cosf(t1[0] * 0.5f), s1 = sinf(t1[0] * 0.5f);
    const float c2 = cosf(t2[0] * 0.5f), s2 = sinf(t2[0] * 0.5f);
    const float c3 = cosf(t3[0] * 0.5f), s3 = sinf(t3[0] * 0.5f);
    const float c4 = cosf(t4[0] * 0.5f), s4 = sinf(t4[0] * 0.5f);
    const float c5 = cosf(t5[0] * 0.5f), s5 = sinf(t5[0] * 0.5f);
    const float c6 = cosf(t6[0] * 0.5f), s6 = sinf(t6[0] * 0.5f);

    // layer1 = kron(X, X, RY(t1), RY(t2))
    A[r][c] = X_e((r >> 3) & 1, (c >> 3) & 1) * X_e((r >> 2) & 1, (c >> 2) & 1) *
              ry_e(c1, s1, (r >> 1) & 1, (c >> 1) & 1) * ry_e(c2, s2, r & 1, c & 1);
    // layer2 = kron(RY(t3), RY(t4), CNOT)
    Bm[r][c] = ry_e(c3, s3, (r >> 3) & 1, (c >> 3) & 1) *
               ry_e(c4, s4, (r >> 2) & 1, (c >> 2) & 1) * CNOT_e(r & 3, c & 3);
    __syncthreads();
    float acc = 0.f;
    for (int k = 0; k < 16; ++k) acc += Bm[r][k] * A[k][c];   // T1 = L2 @ L1
    Cm[r][c] = acc;
    __syncthreads();

    // layer3 = kron(CNOT, I, RY(t5))
    Bm[r][c] = CNOT_e(r >> 2, c >> 2) * I_e((r >> 1) & 1, (c >> 1) & 1) *
               ry_e(c5, s5, r & 1, c & 1);
    A[r][c] = Cm[r][c];
    __syncthreads();
    acc = 0.f;
    for (int k = 0; k < 16; ++k) acc += Bm[r][k] * A[k][c];   // T2 = L3 @ T1
    Cm[r][c] = acc;
    __syncthreads();

    // layer4 = kron(I, RY(t6), I, I): bits 3,1,0 must match
    Bm[r][c] = (((r ^ c) & 0xB) == 0) ? ry_e(c6, s6, (r >> 2) & 1, (c >> 2) & 1) : 0.f;
    A[r][c] = Cm[r][c];
    __syncthreads();
    acc = 0.f;
    for (int k = 0; k < 16; ++k) acc += Bm[r][k] * A[k][c];   // T3 = L4 @ T2
    Cm[r][c] = acc;
    __syncthreads();

    // layer5 = kron(I, I, CNOT) with row/col bits 1<->2 swapped
    {
        int rr = swap12(r), cc = swap12(c);
        Bm[r][c] = (((rr ^ cc) & 0xC) == 0) ? CNOT_e(rr & 3, cc & 3) : 0.f;
    }
    A[r][c] = Cm[r][c];
    __syncthreads();
    acc = 0.f;
    for (int k = 0; k < 16; ++k) acc += Bm[r][k] * A[k][c];   // U = L5 @ T3
    U[t] = acc;
}

// ---------------- kernel 2: alpha = U @ input_state ----------------
__global__ void __launch_bounds__(256) compute_alpha(const float* __restrict__ U,
                                                     const float* __restrict__ s,
                                                     float* __restrict__ alpha) {
    __shared__ float Us[256];
    Us[threadIdx.x] = U[threadIdx.x];
    __syncthreads();
    const int j = blockIdx.x * blockDim.x + threadIdx.x;
    float sv[16];
#pragma unroll
    for (int m = 0; m < 16; ++m) sv[m] = s[m * BQ + j];
#pragma unroll
    for (int k = 0; k < 16; ++k) {
        float acc = 0.f;
#pragma unroll
        for (int m = 0; m < 16; ++m) acc += Us[k * 16 + m] * sv[m];
        alpha[k * BQ + j] = acc;
    }
}

// ---------------- kernel 3: out = alpha^T @ (M @ alpha), WMMA fp32 ----------------
// M = diag((-1)^(k&1)); folded into the B fragment's odd-k component.
__global__ void __launch_bounds__(256) gram_wmma(const float* __restrict__ alpha,
                                                 float* __restrict__ out) {
    const int lane  = threadIdx.x & 31;
    const int wave  = blockIdx.x * (blockDim.x >> 5) + (threadIdx.x >> 5);
    const int strips = BQ / 64;            // 128 strips of 4 tiles along N
    const int tileRow = wave / strips;     // 0..511
    const int strip   = wave % strips;
    const int i0 = tileRow * 16;
    const int j0 = strip * 64;

    const int m     = lane & 15;
    const int khalf = (lane >> 4) << 1;    // lanes 0-15 -> K pair {0,1}; lanes 16-31 -> {2,3}

    // A fragment: A^T[m][k] = alpha[k][i0+m], per 16x4 fp32 A layout
    v2f a[4];
#pragma unroll
    for (int g = 0; g < 4; ++g) {
        const int k0 = 4 * g + khalf;
        a[g].x = alpha[(size_t)k0 * BQ + i0 + m];
        a[g].y = alpha[(size_t)(k0 + 1) * BQ + i0 + m];
    }

    // C/D layout: VGPR r -> row i0+r (lanes 0-15) / i0+r+8 (lanes 16-31), col = j + (lane&15)
    float* __restrict__ pout = out + (size_t)(i0 + (lane >> 4) * 8) * BQ + (lane & 15);

#pragma unroll
    for (int t = 0; t < 4; ++t) {
        const int jb = j0 + t * 16;
        v8f c = {};
#pragma unroll
        for (int g = 0; g < 4; ++g) {
            const int k0 = 4 * g + khalf;
            v2f b;
            b.x =  alpha[(size_t)k0 * BQ + jb + m];        // even k: M sign = +1
            b.y = -alpha[(size_t)(k0 + 1) * BQ + jb + m];  // odd  k: M sign = -1
            c = __builtin_amdgcn_wmma_f32_16x16x4_f32(
                    false, a[g], false, b, (short)0, c, false, false);
        }
        float* __restrict__ p = pout + jb;
#pragma unroll
        for (int r = 0; r < 8; ++r) p[(size_t)r * BQ] = c[r];
    }
}

// ---------------- launcher ----------------
extern "C" void kernel_launch(void* const* d_in, const int* in_sizes, int n_in,
                              void* d_out, int out_size, void* d_ws, size_t ws_size,
                              hipStream_t stream) {
    const float* s  = (const float*)d_in[0];
    const float* t1 = (const float*)d_in[1];
    const float* t2 = (const float*)d_in[2];
    const float* t3 = (const float*)d_in[3];
    const float* t4 = (const float*)d_in[4];
    const float* t5 = (const float*)d_in[5];
    const float* t6 = (const float*)d_in[6];

    float* U     = (float*)d_ws;        // 256 floats
    float* alpha = U + 256;             // 16 * 8192 floats
    float* out   = (float*)d_out;       // 8192 * 8192 floats

    build_U<<<1, 256, 0, stream>>>(t1, t2, t3, t4, t5, t6, U);
    compute_alpha<<<BQ / 256, 256, 0, stream>>>(U, s, alpha);

    // 512 tile-rows * 128 strips = 65536 waves; 8 waves per 256-thread block
    gram_wmma<<<(512 * 128) / 8, 256, 0, stream>>>(alpha, out);
}